// SPINEModel_26903675142683
// MI455X (gfx1250) — compile-verified
//
#include <hip/hip_runtime.h>
#include <math.h>
#include <stdint.h>

typedef __attribute__((ext_vector_type(2))) float v2f;
typedef __attribute__((ext_vector_type(8))) float v8f;

#define BATCH   8192
#define INPD    300
#define HDIM    1000
#define EPSV    1e-6f
#define RHOSTAR 0.15f
#define TOPK    10

// workspace layout (floats)
#define WS_PSL 0
#define WS_REC 1
#define WS_COL 2            // 1000 column sums of h
#define WS_SY  1002
#define WS_SH  1003
#define WS_DY  1004         // 8192
#define WS_DH  9196         // 8192

#define LDA 20              // LDS row stride (floats): 16B-aligned, non-pow2
#define LDB 20

__global__ __launch_bounds__(256) void init_ws(float* __restrict__ w) {
    int i = blockIdx.x * blockDim.x + threadIdx.x;
    if (i < 1004) w[i] = 0.0f;
}

// CDNA5 async copy: global b128 -> LDS, tracked by ASYNCcnt.
__device__ __forceinline__ void async_cp_b128(const float* g, const float* l) {
    unsigned loff = (unsigned)(unsigned long long)l;  // LDS byte offset (low 32b of generic ptr)
    asm volatile("global_load_async_to_lds_b128 %0, %1, off"
                 :: "v"(loff), "v"(g) : "memory");
}
__device__ __forceinline__ void wait_async0() {
    asm volatile("s_wait_asynccnt 0x0" ::: "memory");
}

// C[M,N] = A[M,K] @ B[N,K]^T + bias, with fused epilogue.
// mode 0: C = clamp(.,0,1) -> h; accumulate psl=sum h(1-h) and colsum[n].
// mode 1: C = out; accumulate rec = sum (out - Y)^2.
__global__ __launch_bounds__(256)
void wmma_gemm(const float* __restrict__ A, const float* __restrict__ Bm,
               const float* __restrict__ bias, float* __restrict__ C,
               const float* __restrict__ Y, float* __restrict__ ws,
               int M, int N, int K, int mode)
{
    __shared__ __align__(16) float sA[2][128 * LDA]; // 128 rows x 16 K, double buffered
    __shared__ __align__(16) float sB[2][64 * LDB];  // 64  rows x 16 K, double buffered
    __shared__ float sCol[64];
    __shared__ float sScalar;

    const int tid  = threadIdx.x;
    const int wave = tid >> 5;
    const int lane = tid & 31;
    const int lg   = lane >> 4;     // half-wave select
    const int lr   = lane & 15;

    const int m0 = blockIdx.x * 128;
    const int n0 = blockIdx.y * 64;

    v8f acc[4];
    #pragma unroll
    for (int t = 0; t < 4; ++t) {
        v8f z = {0.f,0.f,0.f,0.f,0.f,0.f,0.f,0.f};
        acc[t] = z;
    }

    const int r     = tid >> 2;        // 0..63
    const int c4    = (tid & 3) * 4;   // 0,4,8,12
    const int nrow  = n0 + r;          // B source row for this thread
    const bool nok  = (nrow < N);

    // Zero B rows that are never written by async copies (n >= N), both buffers.
    if (!nok) {
        float4 z = make_float4(0.f, 0.f, 0.f, 0.f);
        *(float4*)(&sB[0][r * LDB + c4]) = z;
        *(float4*)(&sB[1][r * LDB + c4]) = z;
    }

    const int nstages = (K + 15) / 16;

    // ---- stage issue: async for full-K stages, sync zero-fill for the tail ----
    auto issue_stage = [&](int k0, int buf) {
        int gk = k0 + c4;
        if (k0 + 16 <= K) {
            // full stage: all K in range; per-lane async b128 scatter into LDS
            #pragma unroll
            for (int p = 0; p < 2; ++p) {
                int row = r + p * 64;
                async_cp_b128(A + (size_t)(m0 + row) * K + gk,
                              &sA[buf][row * LDA + c4]);
            }
            if (nok)
                async_cp_b128(Bm + (size_t)nrow * K + gk,
                              &sB[buf][r * LDB + c4]);
        } else {
            // tail stage: synchronous, zero-fill past K
            #pragma unroll
            for (int p = 0; p < 2; ++p) {
                int row = r + p * 64;
                float4 v = make_float4(0.f, 0.f, 0.f, 0.f);
                if (gk < K) v = *(const float4*)(A + (size_t)(m0 + row) * K + gk);
                *(float4*)(&sA[buf][row * LDA + c4]) = v;
            }
            if (nok) {
                float4 v = make_float4(0.f, 0.f, 0.f, 0.f);
                if (gk < K) v = *(const float4*)(Bm + (size_t)nrow * K + gk);
                *(float4*)(&sB[buf][r * LDB + c4]) = v;
            }
        }
    };

    issue_stage(0, 0);   // preload stage 0 into buffer 0

    for (int s = 0; s < nstages; ++s) {
        const int buf = s & 1;
        wait_async0();        // this wave's copies for buffer `buf` are in LDS
        __syncthreads();      // everyone's copies visible; buf^1 free for reuse

        if (s + 1 < nstages) issue_stage((s + 1) * 16, buf ^ 1);

        const float* cA = &sA[buf][0];
        const float* cB = &sB[buf][0];
        #pragma unroll
        for (int kk = 0; kk < 16; kk += 4) {
            // A frag: lane holds (M=lr, K=kk+2*lg+{0,1})
            v2f a = *(const v2f*)(cA + (wave * 16 + lr) * LDA + kk + 2 * lg);
            #pragma unroll
            for (int nt = 0; nt < 4; ++nt) {
                // B frag: lane holds (N=lr, K=kk+2*lg+{0,1})
                v2f b = *(const v2f*)(cB + (nt * 16 + lr) * LDB + kk + 2 * lg);
                acc[nt] = __builtin_amdgcn_wmma_f32_16x16x4_f32(
                    false, a, false, b, (short)0, acc[nt], false, false);
            }
        }
        __syncthreads();      // done reading `buf` before it is overwritten
    }

    // ---- fused epilogue ----
    if (tid == 0) sScalar = 0.0f;
    if (tid < 64) sCol[tid] = 0.0f;
    __syncthreads();

    float scal = 0.0f;  // psl (mode 0) or rec (mode 1) partial
    #pragma unroll
    for (int nt = 0; nt < 4; ++nt) {
        int n = n0 + nt * 16 + lr;
        float bsv = (n < N) ? bias[n] : 0.0f;
        float colpart = 0.0f;
        #pragma unroll
        for (int v = 0; v < 8; ++v) {
            int m = m0 + wave * 16 + v + 8 * lg;   // C layout: VGPR v -> M = v + 8*lg
            float val = acc[nt][v] + bsv;
            if (mode == 0) {
                val = fminf(fmaxf(val, 0.0f), 1.0f);
                if (n < N) {
                    C[(size_t)m * N + n] = val;
                    scal += val * (1.0f - val);
                    colpart += val;
                }
            } else {
                if (n < N) {
                    C[(size_t)m * N + n] = val;
                    float d = val - Y[(size_t)m * N + n];
                    scal += d * d;
                }
            }
        }
        if (mode == 0) {
            colpart += __shfl_xor(colpart, 16, 32);   // combine half-waves (same n)
            if (lg == 0 && n < N) atomicAdd(&sCol[nt * 16 + lr], colpart);
        }
    }
    #pragma unroll
    for (int off = 16; off > 0; off >>= 1) scal += __shfl_down(scal, off, 32);
    if (lane == 0) atomicAdd(&sScalar, scal);
    __syncthreads();
    if (tid == 0) atomicAdd(&ws[mode == 0 ? WS_PSL : WS_REC], sScalar);
    if (mode == 0 && tid < 64) {
        int n = n0 + tid;
        if (n < N) atomicAdd(&ws[WS_COL + n], sCol[tid]);
    }
}

// d[j] = ||x0 - xj + eps||_2 ; diag -> +inf. One wave per j.
__global__ __launch_bounds__(256)
void dist_row0(const float* __restrict__ X, int D, float* __restrict__ dout)
{
    int wave = threadIdx.x >> 5;
    int lane = threadIdx.x & 31;
    int j = blockIdx.x * 8 + wave;
    float s = 0.0f;
    for (int d = lane; d < D; d += 32) {
        float v = X[d] - X[(size_t)j * D + d] + EPSV;
        s += v * v;
    }
    #pragma unroll
    for (int off = 16; off > 0; off >>= 1) s += __shfl_down(s, off, 32);
    if (lane == 0) dout[j] = (j == 0) ? INFINITY : sqrtf(s);
}

// Sum of the 10 smallest of 8192 values. block 0 -> dy, block 1 -> dh.
__global__ __launch_bounds__(256)
void topk_sum(const float* __restrict__ dy, const float* __restrict__ dh,
              float* __restrict__ ws)
{
    __shared__ float cand[256 * TOPK];
    const float* d = (blockIdx.x == 0) ? dy : dh;
    float best[TOPK];
    #pragma unroll
    for (int q = 0; q < TOPK; ++q) best[q] = INFINITY;
    for (int i = threadIdx.x; i < BATCH; i += 256) {
        float v = d[i];
        if (v < best[TOPK - 1]) {
            int p = TOPK - 1;
            while (p > 0 && best[p - 1] > v) { best[p] = best[p - 1]; --p; }
            best[p] = v;
        }
    }
    #pragma unroll
    for (int q = 0; q < TOPK; ++q) cand[threadIdx.x * TOPK + q] = best[q];
    __syncthreads();
    if (threadIdx.x == 0) {
        float bb[TOPK];
        for (int q = 0; q < TOPK; ++q) bb[q] = INFINITY;
        for (int i = 0; i < 256 * TOPK; ++i) {
            float v = cand[i];
            if (v < bb[TOPK - 1]) {
                int p = TOPK - 1;
                while (p > 0 && bb[p - 1] > v) { bb[p] = bb[p - 1]; --p; }
                bb[p] = v;
            }
        }
        float s = 0.0f;
        for (int q = 0; q < TOPK; ++q) s += bb[q];
        ws[(blockIdx.x == 0) ? WS_SY : WS_SH] = s;
    }
}

__global__ __launch_bounds__(256)
void finalize(const float* __restrict__ ws, float* __restrict__ outScalars)
{
    __shared__ float sred[8];
    float part = 0.0f;
    for (int n = threadIdx.x; n < HDIM; n += 256) {
        float m = ws[WS_COL + n] * (1.0f / BATCH) - RHOSTAR;
        m = fmaxf(m, 0.0f);
        part += m * m;
    }
    int lane = threadIdx.x & 31, wave = threadIdx.x >> 5;
    #pragma unroll
    for (int off = 16; off > 0; off >>= 1) part += __shfl_down(part, off, 32);
    if (lane == 0) sred[wave] = part;
    __syncthreads();
    if (threadIdx.x == 0) {
        float asl_sum = 0.0f;
        for (int w = 0; w < 8; ++w) asl_sum += sred[w];
        float rec = ws[WS_REC] / ((float)BATCH * (float)INPD);
        float psl = ws[WS_PSL] / ((float)BATCH * (float)HDIM);
        float asl = asl_sum / (float)HDIM;
        float loc = fabsf(ws[WS_SY] - ws[WS_SH]);
        outScalars[0] = rec + psl + asl + loc;
        outScalars[1] = rec;
        outScalars[2] = psl;
        outScalars[3] = asl;
        outScalars[4] = loc;
    }
}

extern "C" void kernel_launch(void* const* d_in, const int* in_sizes, int n_in,
                              void* d_out, int out_size, void* d_ws, size_t ws_size,
                              hipStream_t stream)
{
    const float* x  = (const float*)d_in[0];
    const float* y  = (const float*)d_in[1];
    const float* W1 = (const float*)d_in[2];
    const float* b1 = (const float*)d_in[3];
    const float* W2 = (const float*)d_in[4];
    const float* b2 = (const float*)d_in[5];
    (void)in_sizes; (void)n_in; (void)out_size; (void)ws_size;

    float* out  = (float*)d_out;                       // [8192, 300]
    float* h    = out + (size_t)BATCH * INPD;          // [8192, 1000]
    float* scal = h + (size_t)BATCH * HDIM;            // 5 scalars
    float* ws   = (float*)d_ws;

    init_ws<<<dim3((1004 + 255) / 256), dim3(256), 0, stream>>>(ws);

    // h = clamp(x @ W1^T + b1, 0, 1); psl + colsum fused
    wmma_gemm<<<dim3(BATCH / 128, (HDIM + 63) / 64), dim3(256), 0, stream>>>(
        x, W1, b1, h, nullptr, ws, BATCH, HDIM, INPD, 0);

    // out = h @ W2^T + b2; rec fused
    wmma_gemm<<<dim3(BATCH / 128, (INPD + 63) / 64), dim3(256), 0, stream>>>(
        h, W2, b2, out, y, ws, BATCH, INPD, HDIM, 1);

    dist_row0<<<dim3(BATCH / 8), dim3(256), 0, stream>>>(y, INPD, ws + WS_DY);
    dist_row0<<<dim3(BATCH / 8), dim3(256), 0, stream>>>(h, HDIM, ws + WS_DH);

    topk_sum<<<dim3(2), dim3(256), 0, stream>>>(ws + WS_DY, ws + WS_DH, ws);

    finalize<<<dim3(1), dim3(256), 0, stream>>>(ws, scal);
}